// AttentionWithCache_20134806684251
// MI455X (gfx1250) — compile-verified
//
#include <hip/hip_runtime.h>
#include <math.h>

typedef __attribute__((ext_vector_type(2))) float v2f;
typedef __attribute__((ext_vector_type(4))) float v4f;
typedef __attribute__((ext_vector_type(8))) float v8f;
typedef int v4i_t __attribute__((vector_size(4 * sizeof(int))));
typedef __attribute__((address_space(1))) v4i_t* gptr_b128;  // global int4*
typedef __attribute__((address_space(3))) v4i_t* lptr_b128;  // LDS int4*

#define HD 128           // head dim
#define NH 16            // heads
#define TN 16            // new tokens
#define TC 4096          // cached tokens
#define DM 2048          // d_model
#define QKVN (3*DM)      // 6144
#define ROWS (16*TN)     // B*T_new = 256
#define KSPLIT 16        // key splits per (b,h)
#define PART_STRIDE 2080 // 16*128 O + 16 m + 16 s
#define SKP 132          // LDS pitch (floats) for K/V tiles: bank = (4*row+d)%64

#ifndef __has_builtin
#define __has_builtin(x) 0
#endif

#if __has_builtin(__builtin_amdgcn_global_load_async_to_lds_b128)
#define HAVE_ASYNC_LD 1
#else
#define HAVE_ASYNC_LD 0
#endif

#if HAVE_ASYNC_LD
# if __has_builtin(__builtin_amdgcn_s_wait_asynccnt)
#  define WAIT_ASYNC(n) __builtin_amdgcn_s_wait_asynccnt(n)
# else
#  define WAIT_ASYNC(n) asm volatile("s_wait_asynccnt %0" :: "i"(n) : "memory")
# endif
#else
# define WAIT_ASYNC(n) __syncthreads()
#endif

// ---------------------------------------------------------------------------
// Generic fp32 WMMA GEMM: C[M,N] = A[M,K] @ B[K,N], all row-major.
// One wave computes a 16x16 C tile via V_WMMA_F32_16X16X4_F32.
// ---------------------------------------------------------------------------
__global__ __launch_bounds__(32) void wmma_gemm_f32(
    const float* __restrict__ A, const float* __restrict__ B,
    float* __restrict__ C, int M, int N, int K) {
  const int lane = threadIdx.x;
  const int half = lane >> 4;
  const int r16  = lane & 15;
  const int n0 = blockIdx.x * 16;
  const int m0 = blockIdx.y * 16;

  v8f acc = {0.f,0.f,0.f,0.f,0.f,0.f,0.f,0.f};
  const float* Arow = A + (size_t)(m0 + r16) * K + 2 * half;
  const float* Bcol = B + (size_t)(2 * half) * N + n0 + r16;

  #pragma unroll 4
  for (int k0 = 0; k0 < K; k0 += 4) {
    v2f a = *(const v2f*)(Arow + k0);
    v2f b;
    b.x = Bcol[(size_t)k0 * N];
    b.y = Bcol[(size_t)(k0 + 1) * N];
    acc = __builtin_amdgcn_wmma_f32_16x16x4_f32(false, a, false, b,
                                                (short)0, acc, false, false);
  }

  float* Cp = C + (size_t)(m0 + half * 8) * N + n0 + r16;
  #pragma unroll
  for (int r = 0; r < 8; ++r) Cp[(size_t)r * N] = acc[r];
}

// ---------------------------------------------------------------------------
// Stage one 16-row x 128-float K tile and V tile into LDS.
// Per async call: 32 lanes x 16B = one fully coalesced 512B key row.
// ---------------------------------------------------------------------------
__device__ __forceinline__ void stage_tile(const float* __restrict__ gk,
                                           const float* __restrict__ gv,
                                           size_t rstride,
                                           float* lk, float* lv, int lane) {
#if HAVE_ASYNC_LD
  #pragma unroll
  for (int k = 0; k < 16; ++k) {
    __builtin_amdgcn_global_load_async_to_lds_b128(
        (gptr_b128)(gk + (size_t)k * rstride + lane * 4),
        (lptr_b128)(lk + k * SKP + lane * 4), 0, 0);
    __builtin_amdgcn_global_load_async_to_lds_b128(
        (gptr_b128)(gv + (size_t)k * rstride + lane * 4),
        (lptr_b128)(lv + k * SKP + lane * 4), 0, 0);
  }
#else
  #pragma unroll
  for (int k = 0; k < 16; ++k) {
    *(v4f*)(lk + k * SKP + lane * 4) = *(const v4f*)(gk + (size_t)k * rstride + lane * 4);
    *(v4f*)(lv + k * SKP + lane * 4) = *(const v4f*)(gv + (size_t)k * rstride + lane * 4);
  }
#endif
}

// ---------------------------------------------------------------------------
// Flash-attention partial: one wave per (b, h, key-split).
// Async double-buffered LDS staging of K/V chunks overlapped with WMMA work.
// ---------------------------------------------------------------------------
__global__ __launch_bounds__(32) void attn_partial(
    const float* __restrict__ qkv,   // [256, 6144]: Q | K_new | V_new per row
    const float* __restrict__ Kc,    // [B,H,4096,128]
    const float* __restrict__ Vc,    // [B,H,4096,128]
    float* __restrict__ part) {      // [B*H*KSPLIT, PART_STRIDE]
  __shared__ float sK[2][16 * SKP];
  __shared__ float sV[2][16 * SKP];
  __shared__ float sS[16 * 18];
  __shared__ float sAl[16];

  const int lane  = threadIdx.x;
  const int half  = lane >> 4;
  const int r16   = lane & 15;
  const int split = blockIdx.x & (KSPLIT - 1);
  const int bh    = blockIdx.x >> 4;          // b*16 + h
  const int b     = bh >> 4;
  const int h     = bh & 15;
  const float scale = 0.08838834764831845f;   // 1/sqrt(128)
  const float NEG = -__builtin_inff();

  // Q tile resident in registers as 32 pre-scaled A-fragments
  v2f qf[32];
  {
    const float* Qrow = qkv + (size_t)(b * TN + r16) * QKVN + h * HD + 2 * half;
    #pragma unroll
    for (int d4 = 0; d4 < 32; ++d4) {
      v2f q = *(const v2f*)(Qrow + d4 * 4);
      qf[d4].x = q.x * scale;
      qf[d4].y = q.y * scale;
    }
  }

  const v8f vzero = {0.f,0.f,0.f,0.f,0.f,0.f,0.f,0.f};
  v8f o[8];
  #pragma unroll
  for (int t = 0; t < 8; ++t) o[t] = vzero;
  float m_row = NEG, l_row = 0.f;   // valid in lanes 0..15 (row = lane)

  const int nchunk = (split == KSPLIT - 1) ? 17 : 16;

  // chunk -> global row bases + row stride (floats)
  auto chunk_ptrs = [&](int c, const float*& gk, const float*& gv, size_t& rs) {
    if (c == 16) {                       // 16 new (projected) keys/values
      gk = qkv + (size_t)(b * TN) * QKVN + DM + h * HD;
      gv = gk + DM;
      rs = QKVN;
    } else {
      size_t kbase = (size_t)bh * TC + split * 256 + c * 16;
      gk = Kc + kbase * HD;
      gv = Vc + kbase * HD;
      rs = HD;
    }
  };

  // prologue: stage chunk 0 into buffer 0
  {
    const float* gk; const float* gv; size_t rs;
    chunk_ptrs(0, gk, gv, rs);
    stage_tile(gk, gv, rs, sK[0], sV[0], lane);
  }

  for (int c = 0; c < nchunk; ++c) {
    float* lk = sK[c & 1];
    float* lv = sV[c & 1];

    if (c + 1 < nchunk) {
      const float* gk; const float* gv; size_t rs;
      chunk_ptrs(c + 1, gk, gv, rs);
      stage_tile(gk, gv, rs, sK[(c + 1) & 1], sV[(c + 1) & 1], lane);
      WAIT_ASYNC(32);   // retire current buffer's 32 async ops (in-order)
    } else {
      WAIT_ASYNC(0);
    }
    asm volatile("" ::: "memory");

    const bool isnew = (c == 16);

    // ---- S = (Q*scale) @ K^T over d=128 (32 WMMAs), K from LDS ----
    v8f sfr = vzero;
    #pragma unroll
    for (int d4 = 0; d4 < 32; ++d4) {
      v2f bK = *(const v2f*)(lk + r16 * SKP + d4 * 4 + 2 * half);
      sfr = __builtin_amdgcn_wmma_f32_16x16x4_f32(false, qf[d4], false, bK,
                                                  (short)0, sfr, false, false);
    }

    // spill S tile to LDS (D layout -> row-major)
    #pragma unroll
    for (int r = 0; r < 8; ++r) sS[(r + half * 8) * 18 + r16] = sfr[r];
    __syncthreads();

    // ---- online softmax: lane q handles row q ----
    if (lane < 16) {
      float vals[16];
      float cmax = NEG;
      #pragma unroll
      for (int j = 0; j < 16; ++j) {
        float v = sS[lane * 18 + j];
        if (isnew && j > lane) v = NEG;     // causal mask on new keys
        vals[j] = v;
        cmax = fmaxf(cmax, v);
      }
      float newm  = fmaxf(m_row, cmax);
      float alpha = (m_row == NEG) ? 0.f : __expf(m_row - newm);
      float psum  = 0.f;
      #pragma unroll
      for (int j = 0; j < 16; ++j) {
        float p = (vals[j] == NEG) ? 0.f : __expf(vals[j] - newm);
        sS[lane * 18 + j] = p;
        psum += p;
      }
      l_row = l_row * alpha + psum;
      m_row = newm;
      sAl[lane] = alpha;
    }
    __syncthreads();

    // ---- rescale accumulators by per-row alpha (LDS broadcast) ----
    #pragma unroll
    for (int r = 0; r < 8; ++r) {
      float al = sAl[r + half * 8];
      #pragma unroll
      for (int t = 0; t < 8; ++t) o[t][r] *= al;
    }

    // ---- O += P @ V (32 WMMAs: 4 k-groups x 8 d-tiles), V from LDS ----
    #pragma unroll
    for (int kk = 0; kk < 4; ++kk) {
      v2f a = *(const v2f*)(&sS[r16 * 18 + kk * 4 + 2 * half]);
      const float* vr0 = lv + (size_t)(kk * 4 + 2 * half) * SKP + r16;
      #pragma unroll
      for (int t = 0; t < 8; ++t) {
        v2f bV;
        bV.x = vr0[t * 16];
        bV.y = vr0[SKP + t * 16];
        o[t] = __builtin_amdgcn_wmma_f32_16x16x4_f32(false, a, false, bV,
                                                     (short)0, o[t], false, false);
      }
    }
    __syncthreads();   // sS reused next chunk
  }

  // ---- store partial (unnormalized O, row max m, row sum l) ----
  float* P = part + (size_t)blockIdx.x * PART_STRIDE;
  #pragma unroll
  for (int t = 0; t < 8; ++t) {
    float* Pr = P + (size_t)(half * 8) * HD + t * 16 + r16;
    #pragma unroll
    for (int r = 0; r < 8; ++r) Pr[(size_t)r * HD] = o[t][r];
  }
  if (lane < 16) {
    P[2048 + lane] = m_row;
    P[2064 + lane] = l_row;
  }
}

// ---------------------------------------------------------------------------
// Combine KSPLIT partials per (b,h) with log-sum-exp merge -> context[256,2048]
// ---------------------------------------------------------------------------
__global__ __launch_bounds__(128) void attn_combine(
    const float* __restrict__ part, float* __restrict__ ctx) {
  const int bh = blockIdx.x;     // 0..255
  const int b  = bh >> 4;
  const int h  = bh & 15;
  const int d  = threadIdx.x;    // 0..127
  const float NEG = -__builtin_inff();
  const float* P0 = part + (size_t)bh * KSPLIT * PART_STRIDE;

  for (int q = 0; q < 16; ++q) {
    float M = NEG;
    #pragma unroll
    for (int s = 0; s < KSPLIT; ++s)
      M = fmaxf(M, P0[(size_t)s * PART_STRIDE + 2048 + q]);
    float L = 0.f, acc = 0.f;
    #pragma unroll
    for (int s = 0; s < KSPLIT; ++s) {
      const float* P = P0 + (size_t)s * PART_STRIDE;
      float ms = P[2048 + q];
      float w  = (ms == NEG) ? 0.f : __expf(ms - M);
      L   += w * P[2064 + q];
      acc += w * P[q * HD + d];
    }
    ctx[(size_t)(b * TN + q) * DM + h * HD + d] = acc / L;
  }
}

// ---------------------------------------------------------------------------
extern "C" void kernel_launch(void* const* d_in, const int* in_sizes, int n_in,
                              void* d_out, int out_size, void* d_ws, size_t ws_size,
                              hipStream_t stream) {
  (void)in_sizes; (void)n_in; (void)out_size; (void)ws_size;
  const float* x    = (const float*)d_in[0];   // [16,16,2048]
  const float* Kc   = (const float*)d_in[1];   // [16,16,4096,128]
  const float* Vc   = (const float*)d_in[2];   // [16,16,4096,128]
  const float* Wqkv = (const float*)d_in[3];   // [2048,6144]
  const float* Wout = (const float*)d_in[4];   // [2048,2048]
  float* out = (float*)d_out;                  // [16,16,2048]

  float* qkv  = (float*)d_ws;                              // 256*6144
  float* part = qkv + (size_t)ROWS * QKVN;                 // 4096*2080
  float* ctx  = part + (size_t)(16 * NH * KSPLIT) * PART_STRIDE; // 256*2048

  // 1) QKV projection: [256,2048] @ [2048,6144]
  wmma_gemm_f32<<<dim3(QKVN / 16, ROWS / 16), 32, 0, stream>>>(
      x, Wqkv, qkv, ROWS, QKVN, DM);

  // 2) split-K flash attention partials: (b,h,split) waves
  attn_partial<<<16 * NH * KSPLIT, 32, 0, stream>>>(qkv, Kc, Vc, part);

  // 3) merge partials -> context [256,2048]
  attn_combine<<<16 * NH, 128, 0, stream>>>(part, ctx);

  // 4) output projection: [256,2048] @ [2048,2048]
  wmma_gemm_f32<<<dim3(DM / 16, ROWS / 16), 32, 0, stream>>>(
      ctx, Wout, out, ROWS, DM, DM);
}